// HeadWise_JSD_76003741270582
// MI455X (gfx1250) — compile-verified
//
#include <hip/hip_runtime.h>
#include <hip/hip_bf16.h>

#define NUM_HEADS 12
#define LN2F 0.69314718055994530942f

typedef __attribute__((ext_vector_type(2))) float v2f;
typedef __attribute__((ext_vector_type(4))) float v4f;
typedef __attribute__((ext_vector_type(8))) float v8f;

// Full 32-lane (wave32) sum using V_WMMA_F32_16X16X4_F32.
// A: a0 = s, a1 = 0  ->  A[m][0]=s_m (lanes 0-15), A[m][2]=s_{m+16} (lanes 16-31)
// B: all ones (layout independent)
// D[m][n] = s_m + s_{m+16} = u_m  (constant over n)
// D VGPR r: lanes 0-15 hold u_r, lanes 16-31 hold u_{r+8}.
// Sum of 8 D regs per lane = half-sum; SWAPX16 swizzle + add = full wave sum.
__device__ __forceinline__ float wave_reduce32(float s) {
    v2f a; a[0] = s;    a[1] = 0.0f;
    v2f b; b[0] = 1.0f; b[1] = 1.0f;
    v8f c = {0.f, 0.f, 0.f, 0.f, 0.f, 0.f, 0.f, 0.f};
    v8f d = __builtin_amdgcn_wmma_f32_16x16x4_f32(
        /*neg_a=*/false, a, /*neg_b=*/false, b,
        /*c_mod=*/(short)0, c, /*reuse_a=*/false, /*reuse_b=*/false);
    float half = ((d[0] + d[1]) + (d[2] + d[3])) + ((d[4] + d[5]) + (d[6] + d[7]));
    // SWAPX16: group-of-32 swizzle, and=0x1f, or=0, xor=0x10 -> 0x401F
    float other = __int_as_float(
        __builtin_amdgcn_ds_swizzle(__float_as_int(half), 0x401F));
    return half + other;
}

// Block-wide sum; valid result on threadIdx.x == 0.
// Must be called by all 256 threads unconditionally (WMMA needs EXEC all-ones).
__device__ __forceinline__ float block_reduce256(float s) {
    __shared__ float lds[8];
    float w = wave_reduce32(s);
    int lane = threadIdx.x & 31;
    int wid  = threadIdx.x >> 5;
    if (lane == 0) lds[wid] = w;
    __syncthreads();
    float tot = 0.0f;
    if (threadIdx.x == 0) {
        #pragma unroll
        for (int i = 0; i < 8; ++i) tot += lds[i];
    }
    return tot;
}

__device__ __forceinline__ v4f ldnt(const v4f* __restrict__ ptr) {
    return __builtin_nontemporal_load(ptr);   // read-once stream: TH=NT
}

__device__ __forceinline__ float jsd4(v4f a, v4f b) {
    float t0 = a[0] * (__log2f(a[0]) - __log2f(b[0]));
    float t1 = a[1] * (__log2f(a[1]) - __log2f(b[1]));
    float t2 = a[2] * (__log2f(a[2]) - __log2f(b[2]));
    float t3 = a[3] * (__log2f(a[3]) - __log2f(b[3]));
    return (t0 + t1) + (t2 + t3);
}

// Pass 1: streaming partial sums of  p * (log2(p) - log2(q)).
// n fits in int (50.3M), byte offsets fit in 31 bits -> 32-bit indexing.
__global__ void jsd_partial_kernel(const float* __restrict__ p,
                                   const float* __restrict__ q,
                                   float* __restrict__ part,
                                   int n) {
    const int tid    = (int)(blockIdx.x * blockDim.x + threadIdx.x);
    const int stride = (int)(gridDim.x * blockDim.x);
    const int n4     = n >> 2;
    const v4f* __restrict__ p4 = (const v4f*)p;
    const v4f* __restrict__ q4 = (const v4f*)q;

    float s0 = 0.0f, s1 = 0.0f, s2 = 0.0f, s3 = 0.0f;
    int i = tid;
    // Unroll x4 with independent loads: 8 B128 loads (4 KB/wave) in flight
    // before any s_wait_loadcnt -> enough MLP to run at HBM rate.
    for (; i + 3 * stride < n4; i += 4 * stride) {
        v4f a0 = ldnt(p4 + i);
        v4f a1 = ldnt(p4 + i + stride);
        v4f a2 = ldnt(p4 + i + 2 * stride);
        v4f a3 = ldnt(p4 + i + 3 * stride);
        v4f b0 = ldnt(q4 + i);
        v4f b1 = ldnt(q4 + i + stride);
        v4f b2 = ldnt(q4 + i + 2 * stride);
        v4f b3 = ldnt(q4 + i + 3 * stride);
        s0 += jsd4(a0, b0);
        s1 += jsd4(a1, b1);
        s2 += jsd4(a2, b2);
        s3 += jsd4(a3, b3);
    }
    for (; i < n4; i += stride) {
        s0 += jsd4(ldnt(p4 + i), ldnt(q4 + i));
    }
    // scalar tail (none for this problem size, kept for safety)
    for (int j = (n4 << 2) + tid; j < n; j += stride) {
        float a = p[j], b = q[j];
        s0 += a * (__log2f(a) - __log2f(b));
    }
    float s = (s0 + s1) + (s2 + s3);

    float tot = block_reduce256(s);
    if (threadIdx.x == 0) part[blockIdx.x] = tot;
}

// Pass 2: deterministic reduction of block partials + final scale.
// out = sum * ln2 * 0.5 / NUM_HEADS   (kl_pm term is exactly zero:
// m = log(0.5*(p+p)) == log(p) bitwise, so em*(m-log p) == 0)
__global__ void jsd_final_kernel(const float* __restrict__ part,
                                 int nparts,
                                 float* __restrict__ out) {
    float s = 0.0f;
    for (int i = threadIdx.x; i < nparts; i += blockDim.x) s += part[i];
    float tot = block_reduce256(s);
    if (threadIdx.x == 0)
        out[0] = tot * (0.5f * LN2F / (float)NUM_HEADS);
}

extern "C" void kernel_launch(void* const* d_in, const int* in_sizes, int n_in,
                              void* d_out, int out_size, void* d_ws, size_t ws_size,
                              hipStream_t stream) {
    const float* p = (const float*)d_in[0];
    const float* q = (const float*)d_in[1];
    float* out     = (float*)d_out;
    float* part    = (float*)d_ws;

    int n = in_sizes[0];

    // 2048 blocks x 256 threads (8 wave32/block) -> 6 unrolled iterations per
    // thread over 12.58M float4 pairs. Clamp partial count to fit d_ws.
    int blocks = 2048;
    long max_parts = (long)(ws_size / sizeof(float));
    if (max_parts < 1) max_parts = 1;
    if ((long)blocks > max_parts) blocks = (int)max_parts;

    jsd_partial_kernel<<<blocks, 256, 0, stream>>>(p, q, part, n);
    jsd_final_kernel<<<1, 256, 0, stream>>>(part, blocks, out);
}